// SingleSampleLinearLayer_609885356755
// MI455X (gfx1250) — compile-verified
//
#include <hip/hip_runtime.h>
#include <stdint.h>

// out[b,n] = sum_s x[b,n,s] * W[n,s] + bias[n]; B=128, N=1024, S=2048, fp32.
// HBM-bound (~0.5 FLOP/byte): stream x once via CDNA5 async global->LDS copies
// (double-buffered, ASYNCcnt), reduce with V_WMMA_F32_16X16X4_F32 (wave32).

typedef __attribute__((ext_vector_type(2))) float v2f;
typedef __attribute__((ext_vector_type(8))) float v8f;

#define BATCH   128
#define NNEUR   1024
#define SEQ     2048
#define WK      32            // K-chunk per double-buffer stage
#define XSTRIDE (WK + 4)      // 36: 36*m mod 64 hits distinct 4-bank groups ->
                              // conflict-free 16-lane transposed b64 reads

__global__ __launch_bounds__(256)
void neuron_gemv_wmma_async(const float* __restrict__ x,
                            const float* __restrict__ W,
                            const float* __restrict__ bias,
                            float* __restrict__ out) {
    __shared__ float Xs[2][BATCH * XSTRIDE];  // 2 x 18,432 B double buffer
    __shared__ float Ws[SEQ];                 // whole W[n,:] row, 8 KB, loaded once

    const int n    = blockIdx.x;              // one block per neuron
    const int tid  = threadIdx.x;             // 256 threads = 8 wave32s
    const int wave = tid >> 5;
    const int lane = tid & 31;
    const int m    = lane & 15;               // A-matrix row (batch in slice)
    const int kk   = (lane >> 4) << 1;        // K slot: lanes0-15 -> 0, 16-31 -> 2
    const int b0   = wave * 16;

    const size_t bstride = (size_t)NNEUR * SEQ;      // batch stride in x (floats)
    const float* xn = x + (size_t)n * SEQ;
    const float* wn = W + (size_t)n * SEQ;

    // ---- One-time: stage all of W[n,0:2048] into LDS (2 rounds of b128) ----
    #pragma unroll
    for (int r = 0; r < SEQ / (256 * 4); ++r) {
        const int i = (r * 256 + tid) * 4;
        const float4 wv = *(const float4*)(wn + i);
        float* dst = &Ws[i];
        dst[0] = wv.x; dst[1] = wv.y; dst[2] = wv.z; dst[3] = wv.w;
    }

    // ---- Per-thread async-copy geometry (constant across chunks) ----
    // 8 lanes cover one 32-float (128 B) row segment; 4 rounds -> rows +0,32,64,96.
    const int col  = (tid & 7) * 4;
    const int row0 = tid >> 3;
    const uint64_t g0 = (uint64_t)(uintptr_t)(xn + (size_t)row0 * bstride + col);
    const uint32_t l0 = (uint32_t)(uintptr_t)(&Xs[0][row0 * XSTRIDE + col]); // LDS offset = low 32b
    const uint64_t growstep = (uint64_t)32 * bstride * sizeof(float);
    const uint32_t lrowstep = 32 * XSTRIDE * sizeof(float);
    const uint32_t lbufstep = BATCH * XSTRIDE * sizeof(float);

    auto issue_chunk = [&](int k0, int buf) {
        uint64_t g = g0 + (uint64_t)k0 * sizeof(float);
        uint32_t l = l0 + (uint32_t)buf * lbufstep;
        #pragma unroll
        for (int r = 0; r < 4; ++r) {
            // CDNA5 async DMA: global -> LDS, tracked by ASYNCcnt, no VGPR staging.
            asm volatile("global_load_async_to_lds_b128 %0, %1, off"
                         :: "v"(l), "v"(g) : "memory");
            g += growstep;
            l += lrowstep;
        }
    };

    // ---- Prologue: fill buffer 0 ----
    issue_chunk(0, 0);
    asm volatile("s_wait_asynccnt 0x0" ::: "memory");
    __syncthreads();   // also covers the plain-LDS W stores above

    v8f acc = {};      // 16x16 f32 accumulator (all columns identical)
    int buf = 0;

    for (int k0 = 0; k0 < SEQ; k0 += WK) {
        // Kick off next chunk into the other buffer while we compute this one.
        if (k0 + WK < SEQ)
            issue_chunk(k0 + WK, buf ^ 1);

        // 8 WMMAs: A = x[b0:b0+16, k:k+4] (transposed LDS reads, conflict-free),
        // B = w chunk broadcast across all 16 columns (value depends only on K slot).
        #pragma unroll
        for (int k = 0; k < WK; k += 4) {
            const v2f a = *(const v2f*)&Xs[buf][(b0 + m) * XSTRIDE + k + kk];
            v2f b;
            b.x = Ws[k0 + k + kk];
            b.y = Ws[k0 + k + kk + 1];
            acc = __builtin_amdgcn_wmma_f32_16x16x4_f32(
                /*neg_a=*/false, a, /*neg_b=*/false, b,
                /*c_mod=*/(short)0, acc, /*reuse_a=*/false, /*reuse_b=*/false);
        }

        // Wait for our async copies into buf^1, then barrier so (a) everyone is done
        // reading buf, (b) all waves' incoming tile data is visible.
        asm volatile("s_wait_asynccnt 0x0" ::: "memory");
        __syncthreads();
        buf ^= 1;
    }

    // ---- Extract column N=0 of C: lane 0 -> M=0..7, lane 16 -> M=8..15 ----
    const float bn = bias[n];
    if ((lane & 15) == 0) {
        const int mbase = (lane >> 4) * 8;
        #pragma unroll
        for (int v = 0; v < 8; ++v) {
            out[(size_t)(b0 + mbase + v) * NNEUR + n] = acc[v] + bn;
        }
    }
}

extern "C" void kernel_launch(void* const* d_in, const int* in_sizes, int n_in,
                              void* d_out, int out_size, void* d_ws, size_t ws_size,
                              hipStream_t stream) {
    const float* x    = (const float*)d_in[0];   // [B, N, S] fp32
    const float* W    = (const float*)d_in[1];   // [N, S]    fp32
    const float* bias = (const float*)d_in[2];   // [N]       fp32
    float* out        = (float*)d_out;           // [B, N]    fp32

    dim3 grid(NNEUR);   // one block per neuron
    dim3 block(256);    // 8 wave32s, 16 batches each
    hipLaunchKernelGGL(neuron_gemv_wmma_async, grid, block, 0, stream,
                       x, W, bias, out);
}